// DeepEncoder_58076547776777
// MI455X (gfx1250) — compile-verified
//
#include <hip/hip_runtime.h>

// ---------------------------------------------------------------------------
// DeepEncoder (3-layer GCN) for MI455X / gfx1250.
//   GEMMs via V_WMMA_F32_16X16X4_F32 (exact f32; total GEMM work is only
//   ~46 GFLOP so quantizing buys nothing). All shapes compile-time so strides
//   fold into load immediate offsets.
//   SpMM via coalesced per-edge gather + f32 L2 atomics (node matrices are
//   ~102 MB -> resident in the 192 MB L2, so random traffic never hits HBM;
//   only the ~115 MB of edge streams come from HBM at 23.3 TB/s).
//   ReLU fused (compile-time) into the A-operand load of the following GEMM.
// ---------------------------------------------------------------------------

typedef __attribute__((ext_vector_type(2))) float v2f;
typedef __attribute__((ext_vector_type(8))) float v8f;

#define N_NODES 100000
#define N_EDGES 3200000
#define IN_FEAT 512
#define HIDDEN  256
#define OUT_FEAT 128

// ---------------------------------------------------------------------------
// Wave-per-(16 x 64)-tile f32 WMMA GEMM:
//   C[M,N] = (RELU ? relu(A) : A)[M,K] * B[K,N]
// Each wave owns one 16-row strip and FOUR 16-col tiles: one A fragment feeds
// four WMMAs. M % 16 == 0 (100000 = 6250*16), K % 4 == 0, N % 64 == 0.
// K and N are template constants -> B-row strides become immediate offsets.
// Fragment layouts per CDNA5 ISA 7.12.2 (32-bit A 16x4, B 4x16, C/D 16x16):
//   A: lane -> M = lane&15, K = k0 + (lane>>4)*2 + {0,1}
//   B: lane -> N = lane&15, K = k0 + (lane>>4)*2 + {0,1}
//   C: VGPR j -> M = j + (lane<16 ? 0 : 8), N = lane&15
// ---------------------------------------------------------------------------
template <int RELU, int K, int N>
__global__ __launch_bounds__(32)
void gcn_gemm_wmma_f32(const float* __restrict__ A, const float* __restrict__ B,
                       float* __restrict__ C) {
    const int lane = threadIdx.x;            // 0..31, full wave (EXEC all ones)
    const int r    = lane & 15;
    const int kh   = (lane >> 4) << 1;       // 0 (lanes 0-15) / 2 (lanes 16-31)
    const int m0   = blockIdx.x << 4;
    const int n0   = blockIdx.y << 6;        // four 16-wide tiles per wave

    v8f acc0 = {0.f, 0.f, 0.f, 0.f, 0.f, 0.f, 0.f, 0.f};
    v8f acc1 = acc0, acc2 = acc0, acc3 = acc0;

    const float* __restrict__ arow = A + (size_t)(m0 + r) * K + kh;
    const float* __restrict__ bcol = B + (size_t)kh * N + n0 + r;

#pragma unroll 4
    for (int k0 = 0; k0 < K; k0 += 4) {
        v2f a, b0, b1, b2, b3;
        a.x  = arow[k0];
        a.y  = arow[k0 + 1];
        b0.x = bcol[k0 * N];           b0.y = bcol[k0 * N + N];
        b1.x = bcol[k0 * N + 16];      b1.y = bcol[k0 * N + N + 16];
        b2.x = bcol[k0 * N + 32];      b2.y = bcol[k0 * N + N + 32];
        b3.x = bcol[k0 * N + 48];      b3.y = bcol[k0 * N + N + 48];
        if (RELU) {                    // compile-time: no cndmask chain
            a.x = fmaxf(a.x, 0.f);
            a.y = fmaxf(a.y, 0.f);
        }
        // (neg_a, A, neg_b, B, c_mod, C, reuse_a, reuse_b)
        acc0 = __builtin_amdgcn_wmma_f32_16x16x4_f32(
            false, a, false, b0, (short)0, acc0, false, false);
        acc1 = __builtin_amdgcn_wmma_f32_16x16x4_f32(
            false, a, false, b1, (short)0, acc1, false, false);
        acc2 = __builtin_amdgcn_wmma_f32_16x16x4_f32(
            false, a, false, b2, (short)0, acc2, false, false);
        acc3 = __builtin_amdgcn_wmma_f32_16x16x4_f32(
            false, a, false, b3, (short)0, acc3, false, false);
    }

    const int rbase = m0 + ((lane >> 4) << 3);
    float* __restrict__ crow = C + (size_t)rbase * N + n0 + (lane & 15);
#pragma unroll
    for (int j = 0; j < 8; ++j) {
        crow[j * N]      = acc0[j];
        crow[j * N + 16] = acc1[j];
        crow[j * N + 32] = acc2[j];
        crow[j * N + 48] = acc3[j];
    }
}

// ---------------------------------------------------------------------------
// Zero-fill (float4 stores; counts are multiples of 4)
// ---------------------------------------------------------------------------
__global__ __launch_bounds__(256)
void gcn_zero_f32(float4* __restrict__ p, long long n4) {
    long long i = blockIdx.x * (long long)blockDim.x + threadIdx.x;
    if (i < n4) p[i] = make_float4(0.f, 0.f, 0.f, 0.f);
}

// ---------------------------------------------------------------------------
// SpMM scatter:  Y[src[e]] += val[e] * X[dst[e]]   (Y pre-zeroed)
// D/4 consecutive lanes cooperate on one edge -> coalesced 16B gathers and
// coalesced atomic scatter. All random traffic is L2-resident (X,Y ~102 MB).
// LOG_DQ = log2(D/4), compile-time so index math is shifts/ands.
// ---------------------------------------------------------------------------
template <int LOG_DQ, int D>
__global__ __launch_bounds__(256)
void gcn_spmm_edges(const int* __restrict__ src, const int* __restrict__ dst,
                    const float* __restrict__ val, const float* __restrict__ X,
                    float* __restrict__ Y, int E) {
    long long g = blockIdx.x * (long long)blockDim.x + threadIdx.x;
    const long long total = (long long)E << LOG_DQ;
    if (g >= total) return;

    const int e = (int)(g >> LOG_DQ);
    const int c = ((int)g & ((1 << LOG_DQ) - 1)) << 2;   // float offset in row

    const float w = val[e];
    const int   d = dst[e];
    const int   s = src[e];

    const float4 x = *(const float4*)(X + (size_t)d * D + c);
    float* __restrict__ yp = Y + (size_t)s * D + c;

    __hip_atomic_fetch_add(yp + 0, w * x.x, __ATOMIC_RELAXED, __HIP_MEMORY_SCOPE_AGENT);
    __hip_atomic_fetch_add(yp + 1, w * x.y, __ATOMIC_RELAXED, __HIP_MEMORY_SCOPE_AGENT);
    __hip_atomic_fetch_add(yp + 2, w * x.z, __ATOMIC_RELAXED, __HIP_MEMORY_SCOPE_AGENT);
    __hip_atomic_fetch_add(yp + 3, w * x.w, __ATOMIC_RELAXED, __HIP_MEMORY_SCOPE_AGENT);
}

// ---------------------------------------------------------------------------
extern "C" void kernel_launch(void* const* d_in, const int* in_sizes, int n_in,
                              void* d_out, int out_size, void* d_ws, size_t ws_size,
                              hipStream_t stream) {
    const float* feat = (const float*)d_in[0];   // [100000, 512]
    const int*   esrc = (const int*)  d_in[1];   // [3.2M]
    const int*   edst = (const int*)  d_in[2];   // [3.2M]
    const float* eval = (const float*)d_in[3];   // [3.2M]
    const float* W0   = (const float*)d_in[4];   // [512, 256]
    const float* W1   = (const float*)d_in[5];   // [256, 256]
    const float* W2   = (const float*)d_in[6];   // [256, 128]
    float* out = (float*)d_out;                  // [100000, 128]

    // Workspace: two ping-pong node-feature buffers, 100000*256 f32 each.
    float* bufA = (float*)d_ws;                                  // GEMM outputs
    float* bufB = bufA + (size_t)N_NODES * HIDDEN;               // SpMM outputs

    const dim3 wave(32);
    const dim3 blk(256);

    const long long nh4 = (long long)N_NODES * HIDDEN / 4;       // 6.4M float4
    const long long no4 = (long long)N_NODES * OUT_FEAT / 4;     // 3.2M float4
    const int zh_blocks = (int)((nh4 + 255) / 256);
    const int zo_blocks = (int)((no4 + 255) / 256);
    const int sp256_blocks = (int)(((long long)N_EDGES * (HIDDEN / 4)   + 255) / 256);
    const int sp128_blocks = (int)(((long long)N_EDGES * (OUT_FEAT / 4) + 255) / 256);

    // ---- layer 1: H0 = feat @ W0 ; X1 = spmm(H0)  (ReLU deferred to next GEMM)
    gcn_gemm_wmma_f32<0, IN_FEAT, HIDDEN>
        <<<dim3(N_NODES / 16, HIDDEN / 64), wave, 0, stream>>>(feat, W0, bufA);
    gcn_zero_f32<<<zh_blocks, blk, 0, stream>>>((float4*)bufB, nh4);
    gcn_spmm_edges<6, HIDDEN><<<sp256_blocks, blk, 0, stream>>>(
        esrc, edst, eval, bufA, bufB, N_EDGES);

    // ---- layer 2: H1 = relu(X1) @ W1 ; X2 = spmm(H1)
    gcn_gemm_wmma_f32<1, HIDDEN, HIDDEN>
        <<<dim3(N_NODES / 16, HIDDEN / 64), wave, 0, stream>>>(bufB, W1, bufA);
    gcn_zero_f32<<<zh_blocks, blk, 0, stream>>>((float4*)bufB, nh4);
    gcn_spmm_edges<6, HIDDEN><<<sp256_blocks, blk, 0, stream>>>(
        esrc, edst, eval, bufA, bufB, N_EDGES);

    // ---- layer 3: H2 = relu(X2) @ W2 ; out = spmm(H2)   (no final activation)
    gcn_gemm_wmma_f32<1, HIDDEN, OUT_FEAT>
        <<<dim3(N_NODES / 16, OUT_FEAT / 64), wave, 0, stream>>>(bufB, W2, bufA);
    gcn_zero_f32<<<zo_blocks, blk, 0, stream>>>((float4*)out, no4);
    gcn_spmm_edges<5, OUT_FEAT><<<sp128_blocks, blk, 0, stream>>>(
        esrc, edst, eval, bufA, out, N_EDGES);
}